// PocketGraphProcessor_28767690948651
// MI455X (gfx1250) — compile-verified
//
#include <hip/hip_runtime.h>
#include <math.h>

// ---- sizes from the reference ----
#define S_IN   29
#define V_IN   3
#define E_IN   5
#define S_DIM  256
#define V_DIM  16
#define N_GRAPHS 256
#define FEAT   (S_DIM + V_DIM)   // 272

typedef __attribute__((ext_vector_type(2))) float v2f;
typedef __attribute__((ext_vector_type(8))) float v8f;

__device__ __forceinline__ void atomAddF(float* p, float v) {
  __hip_atomic_fetch_add(p, v, __ATOMIC_RELAXED, __HIP_MEMORY_SCOPE_AGENT);
}

// ------------------------------------------------------------------
// zero a float region
__global__ void zero_f(float* __restrict__ p, long long n) {
  long long i = (long long)blockIdx.x * blockDim.x + threadIdx.x;
  if (i < n) p[i] = 0.f;
}

// ------------------------------------------------------------------
// Fold edge linear through Wm1_D:  Wep[i][c] = sum_j We[i][j] * Wm1[528+j][c]
// constc[c] = bm1[c] + sum_j be[j] * Wm1[528+j][c]
__global__ __launch_bounds__(256) void prep_we(
    const float* __restrict__ We, const float* __restrict__ be,
    const float* __restrict__ Wm1, const float* __restrict__ bm1,
    float* __restrict__ Wep, float* __restrict__ constc) {
  int c = threadIdx.x;
  const float* WD = Wm1 + (size_t)(2 * S_DIM + V_DIM) * S_DIM; // rows 528..783
  float acc[E_IN] = {0.f, 0.f, 0.f, 0.f, 0.f};
  float cc = bm1[c];
  for (int j = 0; j < S_DIM; ++j) {
    float w = WD[(size_t)j * S_DIM + c];
    cc += be[j] * w;
#pragma unroll
    for (int i = 0; i < E_IN; ++i) acc[i] += We[i * S_DIM + j] * w;
  }
#pragma unroll
  for (int i = 0; i < E_IN; ++i) Wep[i * S_DIM + c] = acc[i];
  constc[c] = cc;
}

// ------------------------------------------------------------------
// Node embedding: s = node_s@Ws+bs ; v_norm = ||node_v@Wv + bv||.
// Also pool v_norm part of feat and per-graph counts (atomics into gsum/gcnt).
__global__ __launch_bounds__(256) void node_embed(
    const float* __restrict__ node_s, const float* __restrict__ node_v,
    const int* __restrict__ batch,
    const float* __restrict__ Ws, const float* __restrict__ bs,
    const float* __restrict__ Wv, const float* __restrict__ bv,
    float* __restrict__ s, float* __restrict__ v_norm,
    float* __restrict__ gsum, float* __restrict__ gcnt, int N) {
  int n = blockIdx.x;
  if (n >= N) return;
  int t = threadIdx.x;
  const float* xs = node_s + (size_t)n * S_IN;
  float a = bs[t];
#pragma unroll
  for (int i = 0; i < S_IN; ++i) a += xs[i] * Ws[i * S_DIM + t];
  s[(size_t)n * S_DIM + t] = a;

  if (t < V_DIM) {
    const float* xv = node_v + (size_t)n * (V_IN * 3); // [c][x]
    float nv = 0.f;
#pragma unroll
    for (int x = 0; x < 3; ++x) {
      float vv = bv[t];
#pragma unroll
      for (int c = 0; c < V_IN; ++c) vv += xv[c * 3 + x] * Wv[c * V_DIM + t];
      nv += vv * vv;
    }
    float vn = sqrtf(nv);
    v_norm[(size_t)n * V_DIM + t] = vn;
    atomAddF(&gsum[(size_t)batch[n] * FEAT + S_DIM + t], vn);
  }
  if (t == 0) atomAddF(&gcnt[batch[n]], 1.0f);
}

// ------------------------------------------------------------------
// WMMA f32 GEMM: out[M x 256] = [A1 | A2] (M x (K1+K2)) @ W ((K1+K2) x 256)
// One block = one 16-row tile; wave w covers cols [32w, 32w+32) with two
// V_WMMA_F32_16X16X4_F32 accumulators. A-frag layout per ISA: lanes 0-15
// hold K=kb,kb+1 (v0,v1), lanes 16-31 hold K=kb+2,kb+3.
__global__ __launch_bounds__(256) void gemm_node_wmma(
    const float* __restrict__ A1, int K1,
    const float* __restrict__ A2, int K2,
    const float* __restrict__ W,
    float* __restrict__ out, int M) {
  int wave = threadIdx.x >> 5;
  int lane = threadIdx.x & 31;
  int halfk = (lane >> 4) * 2;     // 0 or 2
  int l15 = lane & 15;
  int m0 = blockIdx.x * 16;
  int row = m0 + l15;
  if (row >= M) row = M - 1;       // clamp loads; stores guarded below
  int Kt = K1 + K2;
  int n0 = wave * 32;
  v8f c0 = {}; v8f c1 = {};
  for (int kb = 0; kb < Kt; kb += 4) {
    int k = kb + halfk;            // uniform region select: K1 % 4 == 0
    v2f a;
    if (k < K1) {
      const float* p = A1 + (size_t)row * K1 + k;
      a.x = p[0]; a.y = p[1];
    } else {
      const float* p = A2 + (size_t)row * K2 + (k - K1);
      a.x = p[0]; a.y = p[1];
    }
    const float* w0 = W + (size_t)k * S_DIM;
    v2f b0, b1;
    b0.x = w0[n0 + l15];          b0.y = w0[S_DIM + n0 + l15];
    b1.x = w0[n0 + 16 + l15];     b1.y = w0[S_DIM + n0 + 16 + l15];
    c0 = __builtin_amdgcn_wmma_f32_16x16x4_f32(false, a, false, b0, (short)0, c0, false, false);
    c1 = __builtin_amdgcn_wmma_f32_16x16x4_f32(false, a, false, b1, (short)0, c1, false, false);
  }
  // C/D layout: VGPR r -> M = m0 + r (lanes 0-15) or m0 + r + 8 (lanes 16-31)
  int srow = m0 + ((lane >> 4) ? 8 : 0);
  int scol = n0 + l15;
#pragma unroll
  for (int r = 0; r < 8; ++r) {
    int rr = srow + r;
    if (rr < M) {
      out[(size_t)rr * S_DIM + scol]      = c0[r];
      out[(size_t)rr * S_DIM + scol + 16] = c1[r];
    }
  }
}

// ------------------------------------------------------------------
// Per-edge: pre = t_dst[dst] + t_src[src] + edge_s@Wep + constc ; relu ;
// atomic scatter-add into acc[dst], count edges per dst.
// One wave per edge; lane owns 8 contiguous channels (two b128 loads).
// Wep/constc tiles are lane-invariant across edges -> hoisted to VGPRs.
// Next edge's gather rows are prefetched (global_prefetch_b8) one iter ahead.
__global__ __launch_bounds__(256) void edge_kernel(
    const int* __restrict__ edge_index, int E,
    const float* __restrict__ edge_s,
    const float* __restrict__ t_dst, const float* __restrict__ t_src,
    const float* __restrict__ Wep, const float* __restrict__ constc,
    float* __restrict__ acc, float* __restrict__ cnt) {
  int lane = threadIdx.x & 31;
  int wave = (blockIdx.x * blockDim.x + threadIdx.x) >> 5;
  int nwaves = (gridDim.x * blockDim.x) >> 5;
  int c0 = lane * 8;

  // hoist lane's 8-channel slice of Wep and constc into registers
  float wep[E_IN][8];
  float cst[8];
#pragma unroll
  for (int j = 0; j < 8; ++j) {
    cst[j] = constc[c0 + j];
#pragma unroll
    for (int i = 0; i < E_IN; ++i) wep[i][j] = Wep[i * S_DIM + c0 + j];
  }

  int e = wave;
  int sidx = 0, didx = 0;
  if (e < E) {
    sidx = edge_index[e];          // edge_index[0][e] = src (j)
    didx = edge_index[E + e];      // edge_index[1][e] = dst (i)
  }
  for (; e < E; e += nwaves) {
    // pipeline: fetch next edge's indices and prefetch its gather rows
    int en = e + nwaves;
    int sidx_n = sidx, didx_n = didx;
    if (en < E) {
      sidx_n = edge_index[en];
      didx_n = edge_index[E + en];
      __builtin_prefetch(t_src + (size_t)sidx_n * S_DIM + c0, 0, 3);
      __builtin_prefetch(t_dst + (size_t)didx_n * S_DIM + c0, 0, 3);
    }

    float es[E_IN];
#pragma unroll
    for (int i = 0; i < E_IN; ++i) es[i] = edge_s[(size_t)e * E_IN + i];
    const float4* td = (const float4*)(t_dst + (size_t)didx * S_DIM + c0);
    const float4* ts = (const float4*)(t_src + (size_t)sidx * S_DIM + c0);
    float4 d0 = td[0], d1 = td[1];
    float4 s0 = ts[0], s1 = ts[1];
    float pre[8] = {d0.x + s0.x, d0.y + s0.y, d0.z + s0.z, d0.w + s0.w,
                    d1.x + s1.x, d1.y + s1.y, d1.z + s1.z, d1.w + s1.w};
#pragma unroll
    for (int j = 0; j < 8; ++j) {
      float p = pre[j] + cst[j];
#pragma unroll
      for (int i = 0; i < E_IN; ++i) p += es[i] * wep[i][j];
      p = fmaxf(p, 0.f);
      atomAddF(&acc[(size_t)didx * S_DIM + c0 + j], p);
    }
    if (lane == 0) atomAddF(&cnt[didx], 1.0f);

    sidx = sidx_n; didx = didx_n;
  }
}

// ------------------------------------------------------------------
__global__ void finalize_cnt(const float* __restrict__ cnt,
                             float* __restrict__ invc, float* __restrict__ maskb, int N) {
  int i = blockIdx.x * blockDim.x + threadIdx.x;
  if (i < N) {
    float c = cnt[i];
    invc[i]  = c > 0.f ? 1.0f / c : 0.f;
    maskb[i] = c > 0.f ? 1.0f : 0.f;
  }
}

// ------------------------------------------------------------------
// s_new = s + (acc*invc) @ Wm2 + mask*bm2 ; pool s_new into gsum (atomics).
// Same WMMA structure as gemm_node_wmma, K = 256.
__global__ __launch_bounds__(256) void update_pool_wmma(
    const float* __restrict__ acc, const float* __restrict__ invc,
    const float* __restrict__ maskb, const float* __restrict__ s,
    const float* __restrict__ Wm2, const float* __restrict__ bm2,
    const int* __restrict__ batch, float* __restrict__ gsum, int M) {
  int wave = threadIdx.x >> 5;
  int lane = threadIdx.x & 31;
  int halfk = (lane >> 4) * 2;
  int l15 = lane & 15;
  int m0 = blockIdx.x * 16;
  int row = m0 + l15;
  if (row >= M) row = M - 1;
  float ic = invc[row];
  int n0 = wave * 32;
  v8f c0 = {}; v8f c1 = {};
  for (int kb = 0; kb < S_DIM; kb += 4) {
    int k = kb + halfk;
    const float* p = acc + (size_t)row * S_DIM + k;
    v2f a; a.x = p[0] * ic; a.y = p[1] * ic;
    const float* w0 = Wm2 + (size_t)k * S_DIM;
    v2f b0, b1;
    b0.x = w0[n0 + l15];          b0.y = w0[S_DIM + n0 + l15];
    b1.x = w0[n0 + 16 + l15];     b1.y = w0[S_DIM + n0 + 16 + l15];
    c0 = __builtin_amdgcn_wmma_f32_16x16x4_f32(false, a, false, b0, (short)0, c0, false, false);
    c1 = __builtin_amdgcn_wmma_f32_16x16x4_f32(false, a, false, b1, (short)0, c1, false, false);
  }
  int srow = m0 + ((lane >> 4) ? 8 : 0);
  int scol = n0 + l15;
#pragma unroll
  for (int r = 0; r < 8; ++r) {
    int rr = srow + r;
    if (rr < M) {
      float mk = maskb[rr];
      int g = batch[rr];
      float y0 = c0[r] + s[(size_t)rr * S_DIM + scol]      + mk * bm2[scol];
      float y1 = c1[r] + s[(size_t)rr * S_DIM + scol + 16] + mk * bm2[scol + 16];
      atomAddF(&gsum[(size_t)g * FEAT + scol],      y0);
      atomAddF(&gsum[(size_t)g * FEAT + scol + 16], y1);
    }
  }
}

// ------------------------------------------------------------------
// Per-graph: g = gsum/max(gcnt,1) ; o = g@Wo+bo ; layernorm ; relu.
__global__ __launch_bounds__(256) void graph_out(
    const float* __restrict__ gsum, const float* __restrict__ gcnt,
    const float* __restrict__ Wo, const float* __restrict__ bo,
    const float* __restrict__ gamma, const float* __restrict__ beta,
    float* __restrict__ out) {
  __shared__ float gbuf[FEAT];
  __shared__ float rs[S_DIM], rq[S_DIM];
  int g = blockIdx.x, t = threadIdx.x;
  float icnt = 1.0f / fmaxf(gcnt[g], 1.0f);
  gbuf[t] = gsum[(size_t)g * FEAT + t] * icnt;
  if (t < FEAT - S_DIM) gbuf[S_DIM + t] = gsum[(size_t)g * FEAT + S_DIM + t] * icnt;
  __syncthreads();
  float o = bo[t];
  for (int k = 0; k < FEAT; ++k) o += gbuf[k] * Wo[(size_t)k * S_DIM + t];
  rs[t] = o; rq[t] = o * o;
  __syncthreads();
  for (int st = 128; st > 0; st >>= 1) {
    if (t < st) { rs[t] += rs[t + st]; rq[t] += rq[t + st]; }
    __syncthreads();
  }
  float mu = rs[0] * (1.0f / S_DIM);
  float var = rq[0] * (1.0f / S_DIM) - mu * mu;
  float y = (o - mu) * rsqrtf(var + 1e-5f) * gamma[t] + beta[t];
  out[(size_t)g * S_DIM + t] = fmaxf(y, 0.f);
}

// ------------------------------------------------------------------
extern "C" void kernel_launch(void* const* d_in, const int* in_sizes, int n_in,
                              void* d_out, int out_size, void* d_ws, size_t ws_size,
                              hipStream_t stream) {
  const float* node_s = (const float*)d_in[0];
  const float* node_v = (const float*)d_in[1];
  const int*   edge_index = (const int*)d_in[2];
  const float* edge_s = (const float*)d_in[3];
  const int*   batch  = (const int*)d_in[4];
  const float* Ws  = (const float*)d_in[5];
  const float* bs  = (const float*)d_in[6];
  const float* Wv  = (const float*)d_in[7];
  const float* bv  = (const float*)d_in[8];
  const float* We  = (const float*)d_in[9];
  const float* be  = (const float*)d_in[10];
  const float* Wm1 = (const float*)d_in[11];
  const float* bm1 = (const float*)d_in[12];
  const float* Wm2 = (const float*)d_in[13];
  const float* bm2 = (const float*)d_in[14];
  const float* Wo  = (const float*)d_in[15];
  const float* bo  = (const float*)d_in[16];
  const float* gamma = (const float*)d_in[17];
  const float* beta  = (const float*)d_in[18];

  const int N = in_sizes[0] / S_IN;
  const int E = in_sizes[3] / E_IN;

  // workspace layout (floats)
  float* ws = (float*)d_ws;
  size_t off = 0;
  float* s      = ws + off; off += (size_t)N * S_DIM;
  float* v_norm = ws + off; off += (size_t)N * V_DIM;
  float* t_dst  = ws + off; off += (size_t)N * S_DIM;
  float* t_src  = ws + off; off += (size_t)N * S_DIM;
  float* invc   = ws + off; off += N;
  float* maskb  = ws + off; off += N;
  float* Wep    = ws + off; off += E_IN * S_DIM;
  float* constc = ws + off; off += S_DIM;
  // contiguous zero region:
  float* zbase  = ws + off;
  float* acc    = ws + off; off += (size_t)N * S_DIM;
  float* cnt    = ws + off; off += N;
  float* gsum   = ws + off; off += (size_t)N_GRAPHS * FEAT;
  float* gcnt   = ws + off; off += N_GRAPHS;
  long long zcount = (long long)(ws + off - zbase);

  const int mtiles = (N + 15) / 16;

  zero_f<<<(int)((zcount + 255) / 256), 256, 0, stream>>>(zbase, zcount);
  prep_we<<<1, 256, 0, stream>>>(We, be, Wm1, bm1, Wep, constc);
  node_embed<<<N, 256, 0, stream>>>(node_s, node_v, batch, Ws, bs, Wv, bv,
                                    s, v_norm, gsum, gcnt, N);
  // t_dst = s @ Wm1[0:256]
  gemm_node_wmma<<<mtiles, 256, 0, stream>>>(s, S_DIM, s, 0, Wm1, t_dst, N);
  // t_src = [s | v_norm] @ Wm1[256:528]
  gemm_node_wmma<<<mtiles, 256, 0, stream>>>(s, S_DIM, v_norm, V_DIM,
                                             Wm1 + (size_t)S_DIM * S_DIM, t_src, N);
  edge_kernel<<<4096, 256, 0, stream>>>(edge_index, E, edge_s, t_dst, t_src,
                                        Wep, constc, acc, cnt);
  finalize_cnt<<<(N + 255) / 256, 256, 0, stream>>>(cnt, invc, maskb, N);
  update_pool_wmma<<<mtiles, 256, 0, stream>>>(acc, invc, maskb, s, Wm2, bm2,
                                               batch, gsum, N);
  graph_out<<<N_GRAPHS, 256, 0, stream>>>(gsum, gcnt, Wo, bo, gamma, beta,
                                          (float*)d_out);
}